// BlockLinear_77635828842763
// MI455X (gfx1250) — compile-verified
//
#include <hip/hip_runtime.h>
#include <hip/hip_bf16.h>

typedef __attribute__((ext_vector_type(2))) float v2f;
typedef __attribute__((ext_vector_type(8))) float v8f;

// Problem constants
#define N_TOKENS   16384          // B*T
#define IN_F       2048
#define OUT_F      2048
#define KK_STEPS   64             // 256 / 4  (WMMA f32 K=4)
#define N_TILES    16             // 256 / 16
#define NT_GROUP   8              // N-tiles per accumulation pass (64 acc VGPRs)
#define WAVES_PER_BLOCK 8
#define LDS_W_FLOATS   (KK_STEPS * N_TILES * 64)   // 65536 floats = 256 KiB
#define LDS_BIAS_BASE  LDS_W_FLOATS                // bias staged after W fragments
#define LDS_BIAS_FLOATS (N_TILES * 16 * 8)         // 2048 floats = 8 KiB

// Out''(131072 x 256) = X''(131072 x 256) * W''(256 x 256), row=(token,m)
__global__ void __launch_bounds__(256, 1)
block_linear_wmma_f32(const float* __restrict__ x,
                      const float* __restrict__ w,
                      const float* __restrict__ bias,
                      float* __restrict__ out) {
    extern __shared__ float lds[];
    const int tid  = threadIdx.x;            // 0..255
    const int lane = tid & 31;
    const int wave = tid >> 5;

    // ---------------- Stage W'' into LDS in B-fragment order ----------------
    // fragment f = kk*16 + nt ; per-lane slot l holds float2 {B[K0][Nn], B[K0+1][Nn]}
    // with K0 = 4*kk + 2*(l>>4), Nn = 16*nt + (l&15); K = c*16+k, Nn = o*32+n.
    for (int i = tid; i < LDS_W_FLOATS / 2; i += 256) {
        const int f  = i >> 5;            // fragment index
        const int l  = i & 31;            // lane slot within fragment
        const int kk = f >> 4;
        const int nt = f & 15;
        const int K0 = 4 * kk + 2 * (l >> 4);     // even -> (K0, K0+1) same c-block
        const int Nn = 16 * nt + (l & 15);
        const int c  = K0 >> 4, k = K0 & 15;
        const int o  = Nn >> 5, n = Nn & 31;
        const v2f wv = *(const v2f*)&w[(((o * 16 + c) * 32 + n) << 4) + k];
        ((v2f*)lds)[i] = wv;
    }
    // ---------------- Stage bias in D-fragment order ------------------------
    // lds_bias[nt*128 + l15*8 + v] = bias[o*256 + v*32 + n], Ng = 16*nt + l15.
    for (int j = tid; j < LDS_BIAS_FLOATS; j += 256) {
        const int ntg = j >> 7;
        const int l   = (j >> 3) & 15;
        const int v   = j & 7;
        const int Ng  = 16 * ntg + l;
        const int o   = Ng >> 5, n = Ng & 31;
        lds[LDS_BIAS_BASE + j] = bias[o * 256 + v * 32 + n];
    }
    __syncthreads();

    // ---------------- Compute: one 16-row tile per wave ----------------------
    const int rt   = blockIdx.x * WAVES_PER_BLOCK + wave;   // row tile 0..8191
    const int h    = lane >> 4;           // lane half (K phase / token select)
    const int l15  = lane & 15;

    // A fragment source: row = 16*rt + l15 -> token = 2*rt + (l15>>3), m = l15&7
    const long tok_a = 2 * rt + (l15 >> 3);
    const int  m_a   = l15 & 7;
    const float* xrow = x + tok_a * (long)IN_F + m_a * 16 + 2 * h;

    const v2f* bfr = (const v2f*)lds;
    const long outbase = (2 * (long)rt + h) * (long)OUT_F;

    // Two passes of 8 N-tiles: live accumulators stay at 64 VGPRs.
#pragma unroll 1
    for (int g = 0; g < 2; ++g) {
        // Seed accumulators with bias (D = A*B + C) -> epilogue is pure stores.
        v8f acc[NT_GROUP];
#pragma unroll
        for (int nt = 0; nt < NT_GROUP; ++nt)
            acc[nt] = *(const v8f*)&lds[LDS_BIAS_BASE + (g * NT_GROUP + nt) * 128 + l15 * 8];

        // Software-pipelined over c-blocks: 4 k-steps per block, A loads at
        // immediate offsets {0,4,8,12}, next block prefetched before compute.
        const float* xa = xrow;
        v2f a0 = *(const v2f*)(xa + 0);
        v2f a1 = *(const v2f*)(xa + 4);
        v2f a2 = *(const v2f*)(xa + 8);
        v2f a3 = *(const v2f*)(xa + 12);

#pragma unroll 1
        for (int cb = 0; cb < 16; ++cb) {
            const float* xn = xa + (cb < 15 ? 128 : 0);   // safe re-read on last
            const v2f p0 = *(const v2f*)(xn + 0);
            const v2f p1 = *(const v2f*)(xn + 4);
            const v2f p2 = *(const v2f*)(xn + 8);
            const v2f p3 = *(const v2f*)(xn + 12);

#pragma unroll
            for (int q = 0; q < 4; ++q) {
                const v2f aq = (q == 0) ? a0 : (q == 1) ? a1 : (q == 2) ? a2 : a3;
                const int fbase = ((4 * cb + q) * N_TILES + g * NT_GROUP) * 32;
#pragma unroll
                for (int nt = 0; nt < NT_GROUP; ++nt) {
                    const v2f b = bfr[fbase + nt * 32 + lane];   // ds_load_b64
                    acc[nt] = __builtin_amdgcn_wmma_f32_16x16x4_f32(
                        false, aq, false, b, (short)0, acc[nt], false, false);
                }
            }
            a0 = p0; a1 = p1; a2 = p2; a3 = p3;
            xa += 128;
        }

        // ---- Epilogue: pure stores (bias already folded into C seed) --------
        // D layout: VGPR v, lane half h -> M = v + 8h ; N col = l15.
#pragma unroll
        for (int nt = 0; nt < NT_GROUP; ++nt) {
            const int Ng = 16 * (g * NT_GROUP + nt) + l15;
            const int o  = Ng >> 5;
            const int n  = Ng & 31;
#pragma unroll
            for (int v = 0; v < 8; ++v)
                out[outbase + o * 256 + v * 32 + n] = acc[nt][v];
        }
    }
}

extern "C" void kernel_launch(void* const* d_in, const int* in_sizes, int n_in,
                              void* d_out, int out_size, void* d_ws, size_t ws_size,
                              hipStream_t stream) {
    const float* x    = (const float*)d_in[0];
    const float* w    = (const float*)d_in[1];
    const float* bias = (const float*)d_in[2];
    float* out        = (float*)d_out;

    const size_t lds_bytes = (LDS_W_FLOATS + LDS_BIAS_FLOATS) * sizeof(float); // 264 KiB
    (void)hipFuncSetAttribute((const void*)block_linear_wmma_f32,
                              hipFuncAttributeMaxDynamicSharedMemorySize,
                              (int)lds_bytes);

    // 131072 rows / 16 rows per wave = 8192 waves; 8 waves per block
    const int blocks = (N_TOKENS * 8) / 16 / WAVES_PER_BLOCK;   // 1024
    block_linear_wmma_f32<<<blocks, 32 * WAVES_PER_BLOCK, lds_bytes, stream>>>(
        x, w, bias, out);
}